// YawningConsecutiveAdjustment_42580305772648
// MI455X (gfx1250) — compile-verified
//
#include <hip/hip_runtime.h>
#include <cstdint>

// ---------------------------------------------------------------------------
// YawningConsecutiveAdjustment for MI455X (gfx1250)
//
// Memory-bound streak counter: 256 MB int32 gestures -> 16384 floats.
// HBM roofline: 256MB / 23.3TB/s ~= 11us. No matrix FLOPs -> no WMMA;
// the CDNA5 paths that matter are:
//   * global_load_async_to_lds_b128 + s_wait_asynccnt  (4-deep DMA pipeline)
//   * wave32 ballot -> uniform words -> SALU run-length math (co-issues
//     with VALU compare + DS reads)
//
// Run counting is fully parallel: a maximal run of length >= K is counted
// once at its start:  startK[t] = !m[t-1] & m[t] & ... & m[t+K-1].
// On packed bits with logarithmic run-AND:
//   y1 = x & (x>>1)        runs >= 2
//   y2 = y1 & (y1>>2)      runs >= 4
//   a7 = y2 & (y2>>3)      runs >= 7
//   np = ~((x<<1) | prevbit)
//   cnt4 += popc32(y2 & np) ; cnt7 += popc32(a7 & np)
// 64-bit container (cur | next<<32) gives the 6-bit lookahead for free.
// ---------------------------------------------------------------------------

#define WAVES_PER_BLOCK 8
#define CHUNK_INTS      512                  // ints per chunk per row
#define CHUNK_BYTES     (CHUNK_INTS * 4)     // 2048 B
#define WORDS_PER_CHUNK (CHUNK_INTS / 32)    // 16 ballot words per chunk
#define PIPE_BUFS       4                    // async pipeline depth

// One 2KB chunk = 4 per-lane b128 async copies (each instr moves
// 32 lanes x 16B = 512B). ASYNCcnt += 4.
__device__ __forceinline__ void async_issue_chunk(uint32_t ldsByteOff,
                                                  uint64_t gbase,
                                                  uint32_t gByteOff,
                                                  int lane) {
#pragma unroll
    for (int i = 0; i < 4; ++i) {
        uint32_t lOff = ldsByteOff + (uint32_t)(i * 512 + lane * 16);
        uint32_t vOff = gByteOff   + (uint32_t)(i * 512 + lane * 16);
        asm volatile("global_load_async_to_lds_b128 %0, %1, %2"
                     :
                     : "v"(lOff), "v"(vOff), "s"(gbase)
                     : "memory");
    }
}

// Wait until the oldest in-flight chunk (4 async ops) has landed, given
// `groupsAfter` younger chunk-groups are still allowed to be outstanding.
__device__ __forceinline__ void async_wait_groups(int groupsAfter) {
    switch (groupsAfter) {
        case 3:  asm volatile("s_wait_asynccnt 12" ::: "memory"); break;
        case 2:  asm volatile("s_wait_asynccnt 8"  ::: "memory"); break;
        case 1:  asm volatile("s_wait_asynccnt 4"  ::: "memory"); break;
        default: asm volatile("s_wait_asynccnt 0"  ::: "memory"); break;
    }
}

__global__ __launch_bounds__(256) void
YawningConsecutiveAdjustment_kernel(const float* __restrict__ din,
                                    const int*   __restrict__ gseq,
                                    float* __restrict__ out,
                                    int B, int T) {
    __shared__ int lds[WAVES_PER_BLOCK * PIPE_BUFS * CHUNK_INTS];  // 64 KB

    const int lane = threadIdx.x & 31;
    const int wv   = threadIdx.x >> 5;
    const int row  = blockIdx.x * WAVES_PER_BLOCK + wv;
    if (row >= B) return;  // whole-wave uniform exit

    int* wbuf = &lds[wv * PIPE_BUFS * CHUNK_INTS];
    const uint32_t ldsBase = (uint32_t)(uintptr_t)wbuf;        // LDS byte offset
    const uint64_t gbase   = (uint64_t)(uintptr_t)gseq;        // SGPR pair base
    const uint32_t rowOff  = (uint32_t)row * (uint32_t)T * 4u; // < 256MB, fits u32

    const int nchunks = T / CHUNK_INTS;

    // prologue: prime the pipeline with up to 3 chunks
    for (int p = 0; p < PIPE_BUFS - 1 && p < nchunks; ++p)
        async_issue_chunk(ldsBase + (uint32_t)p * CHUNK_BYTES, gbase,
                          rowOff + (uint32_t)p * CHUNK_BYTES, lane);

    uint32_t cnt4 = 0, cnt7 = 0;  // maximal runs >=4 / >=7
    uint32_t cur = 0;             // mask word awaiting its lookahead word
    uint32_t pb  = 0;             // bit 31 of the word before `cur` (m[-1]=0)

    for (int c = 0; c < nchunks; ++c) {
        const int pf = c + PIPE_BUFS - 1;  // chunk to prefetch this iteration
        if (pf < nchunks)
            async_issue_chunk(ldsBase + (uint32_t)(pf & (PIPE_BUFS - 1)) * CHUNK_BYTES,
                              gbase, rowOff + (uint32_t)pf * CHUNK_BYTES, lane);
        // in-order async completion: drain down to the younger groups only
        const int younger = (nchunks - 1 - c) < (PIPE_BUFS - 1)
                          ? (nchunks - 1 - c) : (PIPE_BUFS - 1);
        async_wait_groups(younger);

        const int* cbase = wbuf + (c & (PIPE_BUFS - 1)) * CHUNK_INTS;
#pragma unroll
        for (int w = 0; w < WORDS_PER_CHUNK; ++w) {
            const int g = cbase[w * 32 + lane];                 // conflict-free DS read
            const uint32_t nw = __builtin_amdgcn_ballot_w32(g == 2);
            if (c == 0 && w == 0) { cur = nw; continue; }
            // process `cur` with 6-bit lookahead from `nw`, carry bit `pb`
            const uint64_t x  = (uint64_t)cur | ((uint64_t)nw << 32);
            const uint64_t y1 = x  & (x  >> 1);                 // runs >= 2
            const uint64_t y2 = y1 & (y1 >> 2);                 // runs >= 4
            const uint64_t a7 = y2 & (y2 >> 3);                 // runs >= 7
            const uint64_t np = ~((x << 1) | (uint64_t)pb);     // !m[t-1]
            cnt4 += __popc((uint32_t)(y2 & np));
            cnt7 += __popc((uint32_t)(a7 & np));
            pb  = cur >> 31;
            cur = nw;
        }
    }
    {   // flush final word (past the row end is non-yawning)
        const uint64_t x  = (uint64_t)cur;
        const uint64_t y1 = x  & (x  >> 1);
        const uint64_t y2 = y1 & (y1 >> 2);
        const uint64_t a7 = y2 & (y2 >> 3);
        const uint64_t np = ~((x << 1) | (uint64_t)pb);
        cnt4 += __popc((uint32_t)(y2 & np));
        cnt7 += __popc((uint32_t)(a7 & np));
    }

    if (lane == 0) {
        const int H = (int)cnt4, L = (int)cnt7;
        float ha = 0.0f, la = 0.0f;
        if (H >= 2) ha = 0.18f * expf(-0.5f * (float)(H - 2));
        if (L >= 3) la = 0.05f * expf(-0.5f * (float)(L - 3));
        const float adj = fminf(ha + la, 0.35f);
        const float v   = din[row] + adj;
        out[row] = fminf(fmaxf(v, 0.0f), 1.0f);
    }
}

extern "C" void kernel_launch(void* const* d_in, const int* in_sizes, int n_in,
                              void* d_out, int out_size, void* d_ws, size_t ws_size,
                              hipStream_t stream) {
    const float* din  = (const float*)d_in[0];   // [B,1] float32
    const int*   gseq = (const int*)d_in[1];     // [B,T,1] int32
    float*       out  = (float*)d_out;           // [B,1] float32

    const int B = in_sizes[0];
    const int T = in_sizes[1] / B;               // 4096 (multiple of CHUNK_INTS)

    const int blocks = (B + WAVES_PER_BLOCK - 1) / WAVES_PER_BLOCK;
    hipLaunchKernelGGL(YawningConsecutiveAdjustment_kernel,
                       dim3(blocks), dim3(WAVES_PER_BLOCK * 32), 0, stream,
                       din, gseq, out, B, T);
}